// LatentFeature_48988396978342
// MI455X (gfx1250) — compile-verified
//
#include <hip/hip_runtime.h>
#include <stdint.h>

// ---- constants mirrored from the reference ----
#define HASH_P1 73856093u
#define HASH_P2 19349669u
#define HASH_P3 83492791u
#define BUFFER_SZ 10000000
#define RES_F 0.3f
#define TRAVEL_THRESH_F 100.0f
#define KNB 27
#define BLOCK 256
#define TRAVEL_CAP 2048   // LDS mask-LUT capacity (setup uses 128 frames)

typedef float v4f __attribute__((ext_vector_type(4)));
typedef int   v4i __attribute__((ext_vector_type(4)));

#if defined(__AMDGCN__) && __has_builtin(__builtin_amdgcn_global_store_async_from_lds_b128)
#define USE_ASYNC_DRAIN 1
typedef __attribute__((address_space(1))) v4i gv4i;   // global (prints as __device__)
typedef __attribute__((address_space(3))) v4i lv4i;   // LDS (prints as __shared__)
#else
#define USE_ASYNC_DRAIN 0
#endif

__device__ __forceinline__ void wait_asynccnt0() {
#if defined(__AMDGCN__)
#if __has_builtin(__builtin_amdgcn_s_wait_asynccnt)
    __builtin_amdgcn_s_wait_asynccnt(0);
#else
    asm volatile("s_wait_asynccnt 0x0" ::: "memory");
#endif
#endif
}

__global__ __launch_bounds__(BLOCK) void voxel_query_kernel(
    const float* __restrict__ query_xyz,    // [Nq,3]  streamed once (NT loads)
    const float* __restrict__ points,       // [Np,3]  L2-resident gathers (RT)
    const float* __restrict__ travel,       // [nframes]
    const int*   __restrict__ point_index,  // [BUFFER] L2-resident gathers (RT)
    const int*   __restrict__ timestamp,    // [Np]     L2-resident gathers (RT)
    const int*   __restrict__ frame_id_p,   // [1]
    float* __restrict__ out_sdist,          // [Nq*27]
    float* __restrict__ out_idx,            // [Nq*27] (idx as exact float)
    int Nq, int Np, int nframes)
{
    __shared__ __align__(16) float s_sd[BLOCK * KNB];  // 27648 B sdist staging
    __shared__ __align__(16) float s_id[BLOCK * KNB];  // 27648 B idx staging
    __shared__ int s_maskv[TRAVEL_CAP];                // 8192 B: 0 = keep, -1 = mask

    const int  tid    = (int)threadIdx.x;
    const int  q      = (int)blockIdx.x * BLOCK + tid;
    const bool active = q < Nq;

    const int   fid = frame_id_p[0];                   // uniform -> s_load
    const float cur = travel[fid];                     // uniform scalar
    const float md2 = (float)(3.0 * (2.0 * 0.3) * (2.0 * 0.3));

    // Per-frame travel mask computed ONCE (exact reference expression),
    // packed as an OR-able {0,-1} LUT: per-element masking = ds_load + v_or.
    const bool lut = (nframes <= TRAVEL_CAP);
    if (lut) {
        #pragma clang loop unroll(disable)
        for (int i = tid; i < nframes; i += BLOCK)
            s_maskv[i] = (cur - travel[i] >= TRAVEL_THRESH_F) ? -1 : 0;
    }

    // ---- per-query work, amortized over 27 neighbors ----
    float qx = 0.f, qy = 0.f, qz = 0.f;
    if (active) {
        const float* qp = query_xyz + 3u * (unsigned)q;
        qx = __builtin_nontemporal_load(qp + 0);       // fuses to global_load_b96 th:NT
        qy = __builtin_nontemporal_load(qp + 1);
        qz = __builtin_nontemporal_load(qp + 2);
    }
    const int gx = (int)floorf(qx / RES_F);            // precise div matches reference
    const int gy = (int)floorf(qy / RES_F);
    const int gz = (int)floorf(qz / RES_F);

    uint32_t hx[3], hy[3], hz[3];                      // 9 wrapping muls per query
    #pragma unroll
    for (int d = 0; d < 3; ++d) {
        hx[d] = (uint32_t)(gx + d - 1) * HASH_P1;
        hy[d] = (uint32_t)(gy + d - 1) * HASH_P2;
        hz[d] = (uint32_t)(gz + d - 1) * HASH_P3;
    }

    __syncthreads();                                   // mask LUT ready

    // ---- phase A: 27 hash slots (pure VALU, branchless) ----
    int slot[KNB];
    {
        int n = 0;
        #pragma unroll
        for (int i = 0; i < 3; ++i)
            #pragma unroll
            for (int j = 0; j < 3; ++j)
                #pragma unroll
                for (int kk = 0; kk < 3; ++kk, ++n) {
                    const uint32_t hu = hx[i] ^ hy[j] ^ hz[kk];
                    int s = (int)hu % BUFFER_SZ;       // Python-sign mod
                    s += (s >> 31) & BUFFER_SZ;        // branchless non-negative fix
                    slot[n] = s;
                }
    }

    // ---- phase B: 27 independent hash-table gathers (one clause, 27-deep MLP) ----
    int idx[KNB];
    #pragma unroll
    for (int n = 0; n < KNB; ++n) idx[n] = point_index[slot[n]];

    // ---- phase C: fused tail, fully branchless straight-line code ----
    #pragma unroll
    for (int n = 0; n < KNB; ++n) {
        int id = idx[n];
        const unsigned iw = ((unsigned)id < (unsigned)(Np - 1)) ? (unsigned)id
                                                                : (unsigned)(Np - 1); // -1 wraps
        const int ts = timestamp[iw];
        const int m  = lut ? s_maskv[ts]
                           : ((cur - travel[ts] >= TRAVEL_THRESH_F) ? -1 : 0);
        id |= m;                                       // masked -> -1, else unchanged

        const int pc = (id < 0) ? 0 : id;              // clamped: always-gather (no branch)
        const float* pp = points + 3u * (unsigned)pc;
        const float cx = pp[0] - qx;
        const float cy = pp[1] - qy;
        const float cz = pp[2] - qz;
        const float dist = cx * cx + cy * cy + cz * cz;

        const float sd   = (id < 0) ? md2 : dist;
        const int   oidx = (sd >= md2) ? -1 : id;      // sdist keeps value; only idx masked

        s_sd[tid * KNB + n] = sd;                      // stride-27: bank-conflict-free
        s_id[tid * KNB + n] = (float)oidx;             // < 2^24, exact in f32
    }

    __syncthreads();                                   // results staged

    // ---- drain: coalesced NT b128; async LDS->global path when available ----
    const unsigned base   = (unsigned)blockIdx.x * (BLOCK * KNB);
    const unsigned totalE = (unsigned)Nq * KNB;
    const bool fullblk = (base + BLOCK * KNB <= totalE) && ((totalE & 3u) == 0u);
    if (fullblk) {
        v4f* o4sd = (v4f*)(out_sdist + base);
        v4f* o4id = (v4f*)(out_idx + base);
#if USE_ASYNC_DRAIN
        // CDNA5 GLOBAL_STORE_ASYNC_FROM_LDS_B128: LDS -> memory without the
        // VGPR round-trip; tracked on ASYNCcnt (CDNA5 ISA).
        #pragma unroll
        for (int i = 0; i < 6; ++i) {
            const int l = i * BLOCK + tid;
            __builtin_amdgcn_global_store_async_from_lds_b128(
                (gv4i*)(o4sd + l), (lv4i*)&s_sd[4 * l], 0, 0);
            __builtin_amdgcn_global_store_async_from_lds_b128(
                (gv4i*)(o4id + l), (lv4i*)&s_id[4 * l], 0, 0);
        }
        if (tid < (BLOCK * KNB / 4 - 6 * BLOCK)) {     // remaining 192 v4 rows
            const int l = 6 * BLOCK + tid;
            __builtin_amdgcn_global_store_async_from_lds_b128(
                (gv4i*)(o4sd + l), (lv4i*)&s_sd[4 * l], 0, 0);
            __builtin_amdgcn_global_store_async_from_lds_b128(
                (gv4i*)(o4id + l), (lv4i*)&s_id[4 * l], 0, 0);
        }
        wait_asynccnt0();                              // LDS freed only after drain
#else
        const v4f* s4sd = (const v4f*)s_sd;
        const v4f* s4id = (const v4f*)s_id;
        #pragma unroll
        for (int i = 0; i < 6; ++i) {                  // 6*256 v4 = 6144 of 6912 dwords
            const int l = i * BLOCK + tid;
            __builtin_nontemporal_store(s4sd[l], o4sd + l);
            __builtin_nontemporal_store(s4id[l], o4id + l);
        }
        if (tid < (BLOCK * KNB / 4 - 6 * BLOCK)) {     // remaining 192 v4 = 768 dwords
            const int l = 6 * BLOCK + tid;
            __builtin_nontemporal_store(s4sd[l], o4sd + l);
            __builtin_nontemporal_store(s4id[l], o4id + l);
        }
#endif
    } else {                                           // partial last block: guarded b32
        #pragma unroll
        for (int i = 0; i < KNB; ++i) {
            const unsigned lidx = (unsigned)i * BLOCK + (unsigned)tid;
            const unsigned e    = base + lidx;
            if (e < totalE) {
                __builtin_nontemporal_store(s_sd[lidx], out_sdist + e);
                __builtin_nontemporal_store(s_id[lidx], out_idx + e);
            }
        }
    }
}

extern "C" void kernel_launch(void* const* d_in, const int* in_sizes, int n_in,
                              void* d_out, int out_size, void* d_ws, size_t ws_size,
                              hipStream_t stream) {
    (void)n_in; (void)d_ws; (void)ws_size; (void)out_size;
    const float* query_xyz   = (const float*)d_in[0];
    const float* points      = (const float*)d_in[1];
    const float* travel      = (const float*)d_in[2];
    const int*   point_index = (const int*)d_in[3];
    const int*   timestamp   = (const int*)d_in[4];
    const int*   frame_id    = (const int*)d_in[5];

    const int Nq      = in_sizes[0] / 3;
    const int Np      = in_sizes[1] / 3;
    const int nframes = in_sizes[2];

    float* out_sdist = (float*)d_out;
    float* out_idx   = (float*)d_out + (size_t)Nq * KNB;

    const unsigned blocks = ((unsigned)Nq + BLOCK - 1) / BLOCK;

    voxel_query_kernel<<<blocks, BLOCK, 0, stream>>>(
        query_xyz, points, travel, point_index, timestamp, frame_id,
        out_sdist, out_idx, Nq, Np, nframes);
}